// ModelDL_6339371729607
// MI455X (gfx1250) — compile-verified
//
#include <hip/hip_runtime.h>
#include <math.h>

#define BB 4
#define TT 64
#define DD 512
#define HH 8
#define DHH 64
#define NN 196
#define GG (BB*NN)      /* 784 */
#define CIN 768
#define KTOP 44
#define ALPHA_LRELU 0.2f
#define HPAIR 160       /* LDS floats per k-pair row of H (128 data + 32 pad -> disjoint bank halves) */

typedef float v2f __attribute__((ext_vector_type(2)));
typedef float v8f __attribute__((ext_vector_type(8)));

__device__ __forceinline__ float wred_sum(float v){
#pragma unroll
  for (int o = 16; o > 0; o >>= 1) v += __shfl_xor(v, o, 32);
  return v;
}
__device__ __forceinline__ float wred_max(float v){
#pragma unroll
  for (int o = 16; o > 0; o >>= 1) v = fmaxf(v, __shfl_xor(v, o, 32));
  return v;
}

// feature_map [bt, c, p] -> Afm [(b*196+p)*64 + t, c]   (LDS tile transpose)
__global__ void k_transpose_fm(const float* __restrict__ in, float* __restrict__ out){
  __shared__ float tile[32][33];
  const int bt = blockIdx.z; const int b = bt >> 6; const int t = bt & 63;
  const int p0 = blockIdx.x * 32, c0 = blockIdx.y * 32;
  const int tx = threadIdx.x, ty = threadIdx.y;
#pragma unroll
  for (int i = ty; i < 32; i += 8){
    int p = p0 + tx;
    if (p < NN) tile[i][tx] = in[((size_t)bt * CIN + (c0 + i)) * NN + p];
  }
  __syncthreads();
#pragma unroll
  for (int i = ty; i < 32; i += 8){
    int p = p0 + i;
    if (p < NN) out[(((size_t)(b * NN + p)) * TT + t) * CIN + (c0 + tx)] = tile[tx][i];
  }
}

// per-head W [8][512][64] -> Wt [n=hd*64+d][k]  (512x512, [N,K] row-major)
__global__ void k_transpose_w(const float* __restrict__ W, float* __restrict__ Wt){
  const int idx = blockIdx.x * 256 + threadIdx.x;    // over 512*512
  const int n = idx >> 9, k = idx & 511;
  Wt[idx] = W[((size_t)(n >> 6) * DD + k) * DHH + (n & 63)];
}

// C[M,512] = A[M,K] * B  via V_WMMA_F32_16X16X4_F32.
// Workgroup: 256 threads = 8 waves (4 M x 2 N); wave tile 32x64 (8 accumulators).
// Block tile 128x128; grid = (M/128, N/128).  B stored [N,K] row-major: B[k][n] = Bw[n*K+k].
// EPI 0: relu(acc + bias[n]) -> C[row*Ntot+n]
// EPI 1: scatter -> C[((g*8+hd)*64+t)*64+d], row=g*64+t, hd=n>>6, d=n&63
// EPI 2: acc + Xres[row*Ntot+n] + bias[n] -> C
// EPI 3: gelu_exact(acc + bias[n]) -> C
template<int EPI>
__global__ void k_gemm_wmma(const float* __restrict__ A, const float* __restrict__ Bw,
                            const float* __restrict__ bias, const float* __restrict__ Xres,
                            float* __restrict__ Cout, int K, int Ntot)
{
  const int tid  = threadIdx.x;
  const int wave = tid >> 5;
  const int lane = tid & 31;
  const int hh   = lane >> 4;          // lane half
  const int l16  = lane & 15;
  const int mBase = blockIdx.x * 128 + (wave >> 1) * 32;
  const int nBase = blockIdx.y * 128 + (wave & 1) * 64;
  v8f acc[2][4];
#pragma unroll
  for (int mi = 0; mi < 2; ++mi)
#pragma unroll
    for (int j = 0; j < 4; ++j) acc[mi][j] = (v8f){};
  const float* Arow0 = A + (size_t)(mBase + l16) * K;
  const float* Arow1 = A + (size_t)(mBase + 16 + l16) * K;
#pragma unroll 2
  for (int k0 = 0; k0 < K; k0 += 4){
    const int k = k0 + 2 * hh;
    v2f a0 = *(const v2f*)(Arow0 + k);
    v2f a1 = *(const v2f*)(Arow1 + k);
    v2f b[4];
#pragma unroll
    for (int j = 0; j < 4; ++j){
      const int n = nBase + j * 16 + l16;
      b[j] = *(const v2f*)(Bw + (size_t)n * K + k);
    }
#pragma unroll
    for (int j = 0; j < 4; ++j){
      acc[0][j] = __builtin_amdgcn_wmma_f32_16x16x4_f32(false, a0, false, b[j], (short)0, acc[0][j], false, false);
      acc[1][j] = __builtin_amdgcn_wmma_f32_16x16x4_f32(false, a1, false, b[j], (short)0, acc[1][j], false, false);
    }
  }
#pragma unroll
  for (int mi = 0; mi < 2; ++mi){
#pragma unroll
    for (int r = 0; r < 8; ++r){
      const int row = mBase + mi * 16 + r + 8 * hh;
#pragma unroll
      for (int j = 0; j < 4; ++j){
        const int col = nBase + j * 16 + l16;
        float v = acc[mi][j][r];
        if (EPI == 0){
          Cout[(size_t)row * Ntot + col] = fmaxf(v + bias[col], 0.f);
        } else if (EPI == 1){
          const int g = row >> 6, t = row & 63, hd = col >> 6, d = col & 63;
          Cout[(((size_t)g * HH + hd) * TT + t) * DHH + d] = v;
        } else if (EPI == 2){
          const size_t o = (size_t)row * Ntot + col;
          Cout[o] = v + Xres[o] + bias[col];
        } else {
          float x = v + bias[col];
          Cout[(size_t)row * Ntot + col] = 0.5f * x * (1.f + erff(x * 0.70710678118654752f));
        }
      }
    }
  }
}

// mag[row] = ||X[row, 0:512]||  (one wave per row)
__global__ void k_mag(const float* __restrict__ X, float* __restrict__ mag){
  const int row = blockIdx.x * 8 + (threadIdx.x >> 5);
  const int lane = threadIdx.x & 31;
  const float* p = X + (size_t)row * DD;
  float s = 0.f;
#pragma unroll
  for (int i = 0; i < 16; ++i){ float v = p[lane + 32 * i]; s += v * v; }
  s = wred_sum(s);
  if (lane == 0) mag[row] = sqrtf(s);
}

// stable top-K selection by rank (matches lax.top_k tie-break: lower index wins)
__global__ void k_sel(const float* __restrict__ mag, float* __restrict__ sel){
  __shared__ float m[TT];
  const int g = blockIdx.x, t = threadIdx.x;
  m[t] = mag[(size_t)g * TT + t];
  __syncthreads();
  const float mt = m[t];
  int rank = 0;
  for (int j = 0; j < TT; ++j) rank += (m[j] > mt) || (m[j] == mt && j < t);
  sel[(size_t)g * TT + t] = (rank < KTOP) ? 1.f : 0.f;
}

// s1/s2[g,hd,t] = h[g,hd,t,:]·a1/a2[hd,:]  (one wave per idx)
__global__ void k_s1s2(const float* __restrict__ Hm, const float* __restrict__ a1,
                       const float* __restrict__ a2, float* __restrict__ s1,
                       float* __restrict__ s2){
  const int idx = blockIdx.x * 8 + (threadIdx.x >> 5);
  const int lane = threadIdx.x & 31;
  const int hd = (idx >> 6) & 7;
  const float* hp = Hm + (size_t)idx * DHH;
  const float h0 = hp[lane], h1 = hp[lane + 32];
  float v1 = h0 * a1[hd * DHH + lane] + h1 * a1[hd * DHH + lane + 32];
  float v2 = h0 * a2[hd * DHH + lane] + h1 * a2[hd * DHH + lane + 32];
  v1 = wred_sum(v1); v2 = wred_sum(v2);
  if (lane == 0){ s1[idx] = v1; s2[idx] = v2; }
}

// per (g,hd): masked softmax(leaky(s1_i + s2_j)) then attn·h via WMMA; ELU + signed-sqrt epilogue.
// H kept in k-pair-interleaved LDS: ldsH[(k>>1)*HPAIR + n*2 + (k&1)] -> B frags are single b64 loads.
__global__ void k_attn(const float* __restrict__ Hm, const float* __restrict__ s1,
                       const float* __restrict__ s2, const float* __restrict__ sel,
                       float* __restrict__ out){
  __shared__ float ldsH[(TT / 2) * HPAIR];
  __shared__ float ldsA[TT * 65];
  __shared__ float selL[TT], s2L[TT];
  const int gh = blockIdx.x;             // g*8 + hd
  const int g = gh >> 3, hd = gh & 7;
  const int tid = threadIdx.x, wave = tid >> 5, lane = tid & 31;
  const int hh = lane >> 4, l16 = lane & 15;
  const float* Hbase = Hm + (size_t)gh * TT * DHH;
  for (int i = tid; i < TT * DHH; i += 256){
    const int j = i >> 6, d = i & 63;
    ldsH[(j >> 1) * HPAIR + d * 2 + (j & 1)] = Hbase[i];
  }
  if (tid < TT){ selL[tid] = sel[(size_t)g * TT + tid]; s2L[tid] = s2[(size_t)gh * TT + tid]; }
  __syncthreads();
  // phase 1: 8 waves x 8 rows of masked softmax into ldsA
  for (int rr = 0; rr < 8; ++rr){
    const int i = wave * 8 + rr;
    const float s1i  = s1[(size_t)gh * TT + i];
    const float seli = selL[i];
    const int j0 = lane * 2, j1 = j0 + 1;
    float e0 = s1i + s2L[j0]; e0 = (e0 > 0.f) ? e0 : ALPHA_LRELU * e0;
    float e1 = s1i + s2L[j1]; e1 = (e1 > 0.f) ? e1 : ALPHA_LRELU * e1;
    if (!(seli > 0.5f || selL[j0] > 0.5f)) e0 = -9e15f;
    if (!(seli > 0.5f || selL[j1] > 0.5f)) e1 = -9e15f;
    const float mx = wred_max(fmaxf(e0, e1));
    const float p0 = expf(e0 - mx), p1 = expf(e1 - mx);
    const float inv = 1.f / wred_sum(p0 + p1);
    ldsA[i * 65 + j0] = p0 * inv;
    ldsA[i * 65 + j1] = p1 * inv;
  }
  __syncthreads();
  // phase 2: attn(64x64) x h(64x64); wave w owns tiles (mi = w>>1, ni = (w&1)*2 + {0,1})
  const int mi = wave >> 1;
  const int ni0 = (wave & 1) * 2;
  const int mrow  = mi * 16 + l16;
  const int ncol0 = ni0 * 16 + l16;
  const int ncol1 = ncol0 + 16;
  v8f c0 = {}; v8f c1 = {};
#pragma unroll
  for (int k0 = 0; k0 < TT; k0 += 4){
    const int kp = (k0 >> 1) + hh;       // k-pair index for this lane half
    v2f a, b0, b1;
    a.x = ldsA[mrow * 65 + k0 + 2 * hh];
    a.y = ldsA[mrow * 65 + k0 + 2 * hh + 1];
    b0 = *(const v2f*)(ldsH + kp * HPAIR + ncol0 * 2);
    b1 = *(const v2f*)(ldsH + kp * HPAIR + ncol1 * 2);
    c0 = __builtin_amdgcn_wmma_f32_16x16x4_f32(false, a, false, b0, (short)0, c0, false, false);
    c1 = __builtin_amdgcn_wmma_f32_16x16x4_f32(false, a, false, b1, (short)0, c1, false, false);
  }
#pragma unroll
  for (int r = 0; r < 8; ++r){
    const int i = mi * 16 + r + 8 * hh;
#pragma unroll
    for (int s = 0; s < 2; ++s){
      const int d = (s == 0) ? ncol0 : ncol1;
      float v = (s == 0) ? c0[r] : c1[r];
      v = (v > 0.f) ? v : (expf(v) - 1.f);          // ELU
      v = copysignf(sqrtf(fabsf(v)), v);            // signed sqrt
      out[((size_t)g * TT + i) * DD + hd * DHH + d] = v;
    }
  }
}

// normalize over T (axis=1) per (g, channel), clamp 1e-12; in place
__global__ void k_colnorm(float* __restrict__ buf){
  const int idx = blockIdx.x * 256 + threadIdx.x;
  const int g = idx >> 9, c = idx & 511;
  float* p = buf + ((size_t)g * TT) * DD + c;
  float s = 0.f;
  for (int t = 0; t < TT; ++t){ float v = p[(size_t)t * DD]; s += v * v; }
  const float inv = 1.f / fmaxf(sqrtf(s), 1e-12f);
  for (int t = 0; t < TT; ++t) p[(size_t)t * DD] *= inv;
}

// LayerNorm over last dim (512), in place; one wave per row
__global__ void k_layernorm(float* __restrict__ X, const float* __restrict__ gm,
                            const float* __restrict__ bt){
  const int row = blockIdx.x * 8 + (threadIdx.x >> 5);
  const int lane = threadIdx.x & 31;
  float* p = X + (size_t)row * DD;
  float vals[16]; float s = 0.f;
#pragma unroll
  for (int i = 0; i < 16; ++i){ vals[i] = p[lane + 32 * i]; s += vals[i]; }
  const float mean = wred_sum(s) * (1.f / 512.f);
  float vs = 0.f;
#pragma unroll
  for (int i = 0; i < 16; ++i){ float d = vals[i] - mean; vs += d * d; }
  const float var = wred_sum(vs) * (1.f / 512.f);
  const float inv = rsqrtf(var + 1e-5f);
#pragma unroll
  for (int i = 0; i < 16; ++i){
    const int c = lane + 32 * i;
    p[c] = (vals[i] - mean) * inv * gm[c] + bt[c];
  }
}

// ano_map[(b*64+t)*196+p] = Y[(b*196+p)*64+t, :]·post_w + post_b  (one wave each)
__global__ void k_post(const float* __restrict__ Y, const float* __restrict__ pw,
                       const float* __restrict__ pb, float* __restrict__ out){
  const int idx = blockIdx.x * 8 + (threadIdx.x >> 5);
  const int lane = threadIdx.x & 31;
  const int p = idx % NN; const int bt = idx / NN;
  const int b = bt >> 6, t = bt & 63;
  const float* y = Y + (((size_t)(b * NN + p)) * TT + t) * DD;
  float s = 0.f;
#pragma unroll
  for (int i = 0; i < 16; ++i){ const int c = lane + 32 * i; s += y[c] * pw[c]; }
  s = wred_sum(s);
  if (lane == 0) out[idx] = s + pb[0];
}

// logits[row] = h1[row,:]·w2 + b2  (one wave per row)
__global__ void k_cls2(const float* __restrict__ H1, const float* __restrict__ w2,
                       const float* __restrict__ b2, float* __restrict__ out){
  const int row = blockIdx.x * 8 + (threadIdx.x >> 5);
  const int lane = threadIdx.x & 31;
  const float* p = H1 + (size_t)row * DD;
  float s = 0.f;
#pragma unroll
  for (int i = 0; i < 16; ++i){ const int c = lane + 32 * i; s += p[c] * w2[c]; }
  s = wred_sum(s);
  if (lane == 0) out[row] = s + b2[0];
}

extern "C" void kernel_launch(void* const* d_in, const int* in_sizes, int n_in,
                              void* d_out, int out_size, void* d_ws, size_t ws_size,
                              hipStream_t stream){
  (void)in_sizes; (void)n_in; (void)out_size; (void)ws_size;
  const float* x_v    = (const float*)d_in[0];
  const float* fmap   = (const float*)d_in[1];
  const float* pre_w  = (const float*)d_in[2];
  const float* pre_b  = (const float*)d_in[3];
  const float* post_w = (const float*)d_in[4];
  const float* post_b = (const float*)d_in[5];
  const float* tm_W   = (const float*)d_in[6];
  const float* tm_a1  = (const float*)d_in[7];
  const float* tm_a2  = (const float*)d_in[8];
  const float* tm_lw  = (const float*)d_in[9];
  const float* tm_lb  = (const float*)d_in[10];
  const float* tm_g   = (const float*)d_in[11];
  const float* tm_b   = (const float*)d_in[12];
  const float* fm_W   = (const float*)d_in[13];
  const float* fm_a1  = (const float*)d_in[14];
  const float* fm_a2  = (const float*)d_in[15];
  const float* fm_lw  = (const float*)d_in[16];
  const float* fm_lb  = (const float*)d_in[17];
  const float* fm_g   = (const float*)d_in[18];
  const float* fm_b   = (const float*)d_in[19];
  const float* c_w1   = (const float*)d_in[20];
  const float* c_b1   = (const float*)d_in[21];
  const float* c_w2   = (const float*)d_in[22];
  const float* c_b2   = (const float*)d_in[23];
  float* out = (float*)d_out;

  float* ws = (float*)d_ws;
  const size_t S = (size_t)GG * TT * DD;               // 25,690,112 floats
  float* Xfm  = ws;                                    // [G*T, 512]
  float* bufH = ws + S;                                // h / lin-out / Y
  float* buf2 = ws + 2 * S;                            // attn-out
  float* Afm  = bufH;                                  // transpose target [G*T,768] (aliases bufH+buf2, dead after pre-GEMM)
  float* magb = ws + 3 * S;
  float* selb = magb + (size_t)GG * TT;
  float* s1b  = selb + (size_t)GG * TT;
  float* s2b  = s1b + (size_t)GG * HH * TT;
  float* fmWt = s2b + (size_t)GG * HH * TT;            // fm_W transposed [512,512]
  float* tmWt = fmWt + (size_t)DD * DD;                // tm_W transposed [512,512]
  float* h1b  = s1b;                                   // classifier hidden reuses s1 region

  // 0) one-shot weight transposes (per-head [8,512,64] -> [N,K])
  k_transpose_w<<<(DD * DD) / 256, 256, 0, stream>>>(fm_W, fmWt);
  k_transpose_w<<<(DD * DD) / 256, 256, 0, stream>>>(tm_W, tmWt);

  // 1) feature_map layout transpose -> [g, t, c]
  k_transpose_fm<<<dim3(7, 24, BB * TT), dim3(32, 8), 0, stream>>>(fmap, Afm);
  // 2) pre 1x1 conv + bias + ReLU (WMMA f32)
  k_gemm_wmma<0><<<dim3((GG * TT) / 128, 4), 256, 0, stream>>>(Afm, pre_w, pre_b, nullptr, Xfm, CIN, DD);

  // 3) FeatureMapTemporalModule (G = 784)
  k_mag<<<(GG * TT) / 8, 256, 0, stream>>>(Xfm, magb);
  k_sel<<<GG, TT, 0, stream>>>(magb, selb);
  k_gemm_wmma<1><<<dim3((GG * TT) / 128, 4), 256, 0, stream>>>(Xfm, fmWt, nullptr, nullptr, bufH, DD, DD);
  k_s1s2<<<(GG * HH * TT) / 8, 256, 0, stream>>>(bufH, fm_a1, fm_a2, s1b, s2b);
  k_attn<<<GG * HH, 256, 0, stream>>>(bufH, s1b, s2b, selb, buf2);
  k_colnorm<<<(GG * DD) / 256, 256, 0, stream>>>(buf2);
  k_gemm_wmma<2><<<dim3((GG * TT) / 128, 4), 256, 0, stream>>>(buf2, fm_lw, fm_lb, Xfm, bufH, DD, DD);
  k_layernorm<<<(GG * TT) / 8, 256, 0, stream>>>(bufH, fm_g, fm_b);
  // 4) post 1x1 conv -> ano_map (second output)
  k_post<<<(BB * TT * NN) / 8, 256, 0, stream>>>(bufH, post_w, post_b, out + BB * TT);

  // 5) TemporalModule on x_v (G = 4), same kernels/buffers
  k_mag<<<(BB * TT) / 8, 256, 0, stream>>>(x_v, magb);
  k_sel<<<BB, TT, 0, stream>>>(magb, selb);
  k_gemm_wmma<1><<<dim3((BB * TT) / 128, 4), 256, 0, stream>>>(x_v, tmWt, nullptr, nullptr, bufH, DD, DD);
  k_s1s2<<<(BB * HH * TT) / 8, 256, 0, stream>>>(bufH, tm_a1, tm_a2, s1b, s2b);
  k_attn<<<BB * HH, 256, 0, stream>>>(bufH, s1b, s2b, selb, buf2);
  k_colnorm<<<(BB * DD) / 256, 256, 0, stream>>>(buf2);
  k_gemm_wmma<2><<<dim3((BB * TT) / 128, 4), 256, 0, stream>>>(buf2, tm_lw, tm_lb, x_v, bufH, DD, DD);
  k_layernorm<<<(BB * TT) / 8, 256, 0, stream>>>(bufH, tm_g, tm_b);

  // 6) classifier: GEMM+GELU (WMMA) then 512->1 dot -> logits (first output)
  k_gemm_wmma<3><<<dim3((BB * TT) / 128, 4), 256, 0, stream>>>(bufH, c_w1, c_b1, nullptr, h1b, DD, DD);
  k_cls2<<<(BB * TT) / 8, 256, 0, stream>>>(h1b, c_w2, c_b2, out);
}